// UParamLayer_69166153335171
// MI455X (gfx1250) — compile-verified
//
#include <hip/hip_runtime.h>

typedef __attribute__((ext_vector_type(2))) float v2f;
typedef __attribute__((ext_vector_type(8))) float v8f;

constexpr int kDim   = 16;
constexpr int kE     = 120;   // rotations per batch element
constexpr int kWaves = 8;     // waves (batch elements) per block

#define WMMA_F32(A, B, C) \
    __builtin_amdgcn_wmma_f32_16x16x4_f32(false, (A), false, (B), (short)0, (C), false, false)

__global__ __launch_bounds__(kWaves * 32, 2)
void clements_uparam_kernel(const float* __restrict__ params,
                            const float* __restrict__ state_re,
                            const float* __restrict__ state_im,
                            const int*   __restrict__ targets,
                            float*       __restrict__ out,
                            int batch)
{
    __shared__ float sU[kWaves][2][kDim][kDim];   // per-wave scratch: [re/im][row][col]

    const int lane = threadIdx.x & 31;
    const int wave = threadIdx.x >> 5;
    const int b    = blockIdx.x * kWaves + wave;
    if (b >= batch) return;                        // wave-uniform exit

    float (*Ur)[kDim] = sU[wave][0];
    float (*Ui)[kDim] = sU[wave][1];

    const int  col = lane & 15;
    const bool hi  = lane >= 16;

    // ---- U = I ------------------------------------------------------------
    #pragma unroll
    for (int q = 0; q < (kDim * kDim) / 32; ++q) {
        int idx = lane + 32 * q;
        int r = idx >> 4, c = idx & 15;
        Ur[r][c] = (r == c) ? 1.0f : 0.0f;
        Ui[r][c] = 0.0f;
    }

    // ---- sequential Clements rotation chain -------------------------------
    // lane l<16 produces new row m at column `col`; lane l>=16 produces new row n.
    // LDS ops within a wave complete in order -> no barrier needed.
    const float* prm = params + (size_t)b * (2 * kE);
    for (int e = 0; e < kE; ++e) {
        int m = targets[2 * e + 0];
        int n = targets[2 * e + 1];
        float t = prm[e];
        float p = prm[kE + e];
        float st, ct, sp, cp;
        __sincosf(t, &st, &ct);
        __sincosf(p, &sp, &cp);

        float rm_r = Ur[m][col], rm_i = Ui[m][col];
        float rn_r = Ur[n][col], rn_i = Ui[n][col];

        // e^{ip} * (hi ? s : c) as the complex coefficient on row m
        float amp = hi ? st : ct;
        float ar = cp * amp, ai = sp * amp;
        float sgn = hi ? 1.0f : -1.0f;            // + c*rn  vs  - s*rn
        float oth = hi ? ct : st;

        float o_r = ar * rm_r - ai * rm_i + sgn * oth * rn_r;
        float o_i = ar * rm_i + ai * rm_r + sgn * oth * rn_i;

        int row = hi ? n : m;
        Ur[row][col] = o_r;
        Ui[row][col] = o_i;
    }

    // ---- pull U fragments (A-layout; also valid as B-layout of U^T) -------
    // A 16x4 f32: reg0/reg1 = K pair; lanes 0-15 -> K=4k,4k+1 ; lanes 16-31 -> K=4k+2,4k+3
    const int frow = col;                 // M row owned by this lane
    const int koff = hi ? 2 : 0;
    v2f a_ur[4], a_ui[4], a_uin[4];
    #pragma unroll
    for (int k = 0; k < 4; ++k) {
        int c0 = 4 * k + koff;
        a_ur[k]  = v2f{Ur[frow][c0], Ur[frow][c0 + 1]};
        a_ui[k]  = v2f{Ui[frow][c0], Ui[frow][c0 + 1]};
        a_uin[k] = -a_ui[k];
    }

    // ---- load rho as B fragments (B 4x16: reg = K row, lane = N col) ------
    const float* Rr = state_re + (size_t)b * (kDim * kDim);
    const float* Ri = state_im + (size_t)b * (kDim * kDim);
    v2f b_rr[4], b_ri[4];
    #pragma unroll
    for (int k = 0; k < 4; ++k) {
        int r0 = 4 * k + koff;
        b_rr[k] = v2f{Rr[r0 * kDim + col], Rr[(r0 + 1) * kDim + col]};
        b_ri[k] = v2f{Ri[r0 * kDim + col], Ri[(r0 + 1) * kDim + col]};
    }

    // ---- left = U * rho  (complex: 4 real 16x16x16 matmuls) ---------------
    v8f Lr = {}, Li = {};
    #pragma unroll
    for (int k = 0; k < 4; ++k) {
        Lr = WMMA_F32(a_ur[k],  b_rr[k], Lr);   // + Ur*Rr
        Lr = WMMA_F32(a_uin[k], b_ri[k], Lr);   // - Ui*Ri
        Li = WMMA_F32(a_ur[k],  b_ri[k], Li);   // + Ur*Ri
        Li = WMMA_F32(a_ui[k],  b_rr[k], Li);   // + Ui*Rr
    }

    // ---- round-trip `left` through LDS: C/D layout -> A layout ------------
    // C/D: reg v holds row (v + hi*8), col = lane&15.  (U slab is dead now.)
    #pragma unroll
    for (int v = 0; v < 8; ++v) {
        int r = v + (hi ? 8 : 0);
        Ur[r][col] = Lr[v];
        Ui[r][col] = Li[v];
    }
    v2f a_lr[4], a_li[4], a_lrn[4];
    #pragma unroll
    for (int k = 0; k < 4; ++k) {
        int c0 = 4 * k + koff;
        a_lr[k]  = v2f{Ur[frow][c0], Ur[frow][c0 + 1]};
        a_li[k]  = v2f{Ui[frow][c0], Ui[frow][c0 + 1]};
        a_lrn[k] = -a_lr[k];
    }

    // ---- out = left * conj(U)^T -------------------------------------------
    // B-fragments of U^T are register-identical to A-fragments of U.
    // Out_r = Lr*Ur^T + Li*Ui^T ;  Out_i = Li*Ur^T - Lr*Ui^T
    v8f Or = {}, Oi = {};
    #pragma unroll
    for (int k = 0; k < 4; ++k) {
        Or = WMMA_F32(a_lr[k],  a_ur[k], Or);
        Or = WMMA_F32(a_li[k],  a_ui[k], Or);
        Oi = WMMA_F32(a_li[k],  a_ur[k], Oi);
        Oi = WMMA_F32(a_lrn[k], a_ui[k], Oi);
    }

    // ---- store (BATCH, 2, 16, 16) -----------------------------------------
    float* op = out + (size_t)b * (2 * kDim * kDim);
    #pragma unroll
    for (int v = 0; v < 8; ++v) {
        int r = v + (hi ? 8 : 0);
        op[r * kDim + col]                     = Or[v];
        op[kDim * kDim + r * kDim + col]       = Oi[v];
    }
}

extern "C" void kernel_launch(void* const* d_in, const int* in_sizes, int n_in,
                              void* d_out, int out_size, void* d_ws, size_t ws_size,
                              hipStream_t stream) {
    const float* params   = (const float*)d_in[0];
    const float* state_re = (const float*)d_in[1];
    const float* state_im = (const float*)d_in[2];
    const int*   targets  = (const int*)d_in[3];
    float* out = (float*)d_out;

    int batch  = in_sizes[0] / (2 * kE);          // params is (BATCH, 2*E)
    int blocks = (batch + kWaves - 1) / kWaves;
    clements_uparam_kernel<<<blocks, kWaves * 32, 0, stream>>>(
        params, state_re, state_im, targets, out, batch);
}